// LayerWise_CartPool2D_53953379172598
// MI455X (gfx1250) — compile-verified
//
#include <hip/hip_runtime.h>
#include <hip/hip_bf16.h>
#include <stdint.h>

// LayerWise_CartPool2D: 3x3 zero-padded softmax soft-argmax over NHWC f32.
// x: [16,128,128,64] f32  ->  out: [16,128,128,128] f32 (ax ch 0..63, ay ch 64..127)
//
// Roofline: 64MB in + 128MB out = 192MB compulsory -> ~8.2us @ 23.3 TB/s.
// Compute: 151M v_exp_f32. Whole input fits in 192MB L2, so the 9x window
// re-reads stay on-chip (LDS via async copy engine + L2 row overlap).
// WMMA intentionally unused: the only matmul is E[Mx9] x V[9x3] (<6% WMMA tile
// utilization) -- the VALU FMA tree is strictly cheaper.
//
// CDNA5 paths: global_load_async_to_lds_b128 (ASYNCcnt) staging, s_wait_asynccnt,
// wave32 lane layout (2 channels/lane -> ds_load_b64 / global_store_b64).

namespace {
constexpr int Bn = 16;
constexpr int Hn = 128;
constexpr int Wn = 128;
constexpr int Cn = 64;
constexpr int ROW_FLOATS = Wn * Cn;          // 8192 floats = 32 KB per staged row
constexpr int ROW_BYTES  = ROW_FLOATS * 4;   // 32768
constexpr int SMEM_BYTES = 3 * ROW_BYTES;    // 96 KB dynamic LDS
} // namespace

__device__ __forceinline__ void async_g2l_b128(unsigned lds_addr, const void* gaddr) {
  // GLOBAL_LOAD_ASYNC_TO_LDS_B128: per-lane 16B DMA into LDS, tracked by ASYNCcnt.
  asm volatile("global_load_async_to_lds_b128 %0, %1, off"
               :: "v"(lds_addr), "v"(gaddr)
               : "memory");
}

__device__ __forceinline__ void wait_async0() {
#if __has_builtin(__builtin_amdgcn_s_wait_asynccnt)
  __builtin_amdgcn_s_wait_asynccnt(0);
#else
  asm volatile("s_wait_asynccnt 0" ::: "memory");
#endif
}

// Softmax soft-argmax over one 3x3 window (already zero-padded values).
__device__ __forceinline__ void sam9(float a00, float a01, float a02,
                                     float a10, float a11, float a12,
                                     float a20, float a21, float a22,
                                     float wv, float hv,
                                     float& ax, float& ay) {
  const float e00 = __expf(a00), e01 = __expf(a01), e02 = __expf(a02);
  const float e10 = __expf(a10), e11 = __expf(a11), e12 = __expf(a12);
  const float e20 = __expf(a20), e21 = __expf(a21), e22 = __expf(a22);

  const float col1 = e01 + e11 + e21;            // j == 1 taps (wx = 1)
  const float col2 = e02 + e12 + e22;            // j == 2 taps (wx = 2)
  const float row1 = e10 + e11 + e12;            // i == 1 taps (wy = 1)
  const float row2 = e20 + e21 + e22;            // i == 2 taps (wy = 2)
  const float sumE = (e00 + e10 + e20) + col1 + col2;
  const float inv  = __builtin_amdgcn_rcpf(sumE);  // one rcp serves both coords

  const float sumX = col1 + 2.f * col2;
  const float sumY = row1 + 2.f * row2;
  ax = (sumX * inv - 1.f + wv) * (1.f / (float)Wn);
  ay = (sumY * inv - 1.f + hv) * (1.f / (float)Hn);
}

__global__ __launch_bounds__(256)
void cartpool2d_kernel(const float* __restrict__ x, float* __restrict__ out) {
  extern __shared__ float smem[];  // [3][W*C] staged rows h-1, h, h+1
  const int tid = threadIdx.x;
  const int bh  = blockIdx.x;      // 0 .. B*H-1
  const int b   = bh >> 7;         // / 128
  const int h   = bh & 127;

  // Raw 32-bit LDS base of the dynamic allocation (generic ptr low 32 bits).
  const unsigned lbase = (unsigned)(size_t)smem;

  // ---- Stage 3 input rows into LDS via the async copy engine ----
  #pragma unroll
  for (int r = 0; r < 3; ++r) {
    const int hh = h - 1 + r;
    if (hh >= 0 && hh < Hn) {
      const char* g = (const char*)(x + ((size_t)(b * Hn + hh) * Wn) * Cn);
      const unsigned lr = lbase + (unsigned)(r * ROW_BYTES);
      #pragma unroll
      for (int k = 0; k < 8; ++k) {              // 256 thr * 8 * 16B = 32KB
        const unsigned off = (unsigned)((k * 256 + tid) * 16);
        async_g2l_b128(lr + off, g + off);
      }
    } else {
      // Reference zero-pads BEFORE softmax: padded taps contribute exp(0)=1,
      // so halo rows must hold real zeros.
      float4* p = (float4*)(smem + r * ROW_FLOATS);
      const float4 z = make_float4(0.f, 0.f, 0.f, 0.f);
      #pragma unroll
      for (int k = 0; k < 8; ++k) p[k * 256 + tid] = z;
    }
  }
  wait_async0();
  __syncthreads();

  // ---- Compute: each lane owns 2 adjacent channels and a 16-wide w strip ----
  // tid = wave*32 + lane; lane -> channel pair (ds_load_b64 / store_b64 width),
  // wave -> strip (w is wave-uniform, so edge predication stays uniform).
  const int lane  = tid & 31;                    // channel pair index, c2 = 2*lane
  const int strip = tid >> 5;                    // 0..7
  const int w0    = strip * 16;

  const float2* R0 = (const float2*)smem;                     // row h-1
  const float2* R1 = (const float2*)(smem + ROW_FLOATS);      // row h
  const float2* R2 = (const float2*)(smem + 2 * ROW_FLOATS);  // row h+1
  // float2 index for column w: w*(Cn/2) + lane
  auto idx = [&](int w) { return w * (Cn / 2) + lane; };

  // Sliding window columns: p<row><col>, col 0 = w-1, col 1 = w, col 2 = w+1
  float2 p00, p10, p20, p01, p11, p21;
  if (w0 == 0) {
    p00 = p10 = p20 = make_float2(0.f, 0.f);     // global left edge (zero pad)
  } else {
    const int o = idx(w0 - 1);
    p00 = R0[o]; p10 = R1[o]; p20 = R2[o];
  }
  {
    const int o = idx(w0);
    p01 = R0[o]; p11 = R1[o]; p21 = R2[o];
  }

  float* op = out + ((size_t)(b * Hn + h) * Wn) * (2 * Cn);
  const float hv = (float)h;
  const int c2 = lane * 2;

  #pragma unroll 4
  for (int w = w0; w < w0 + 16; ++w) {
    float2 p02, p12, p22;
    if (w + 1 < Wn) {                            // wave-uniform branch
      const int o = idx(w + 1);
      p02 = R0[o]; p12 = R1[o]; p22 = R2[o];     // 3 ds_load_b64 per 2 elems
    } else {
      p02 = p12 = p22 = make_float2(0.f, 0.f);   // global right edge
    }

    const float wv = (float)w;
    float ax0, ay0, ax1, ay1;
    sam9(p00.x, p01.x, p02.x, p10.x, p11.x, p12.x, p20.x, p21.x, p22.x,
         wv, hv, ax0, ay0);
    sam9(p00.y, p01.y, p02.y, p10.y, p11.y, p12.y, p20.y, p21.y, p22.y,
         wv, hv, ax1, ay1);

    // Lanes cover 64 consecutive floats -> 256B per wave per store (coalesced).
    *(float2*)&op[(size_t)w * (2 * Cn) + c2]      = make_float2(ax0, ax1);
    *(float2*)&op[(size_t)w * (2 * Cn) + Cn + c2] = make_float2(ay0, ay1);

    p00 = p01; p01 = p02;                        // slide window
    p10 = p11; p11 = p12;
    p20 = p21; p21 = p22;
  }
}

extern "C" void kernel_launch(void* const* d_in, const int* in_sizes, int n_in,
                              void* d_out, int out_size, void* d_ws, size_t ws_size,
                              hipStream_t stream) {
  (void)in_sizes; (void)n_in; (void)out_size; (void)d_ws; (void)ws_size;
  const float* x = (const float*)d_in[0];
  float* out = (float*)d_out;

  dim3 grid(Bn * Hn);   // 2048 workgroups: one per (batch, output row)
  dim3 block(256);      // 8 wave32s
  cartpool2d_kernel<<<grid, block, SMEM_BYTES, stream>>>(x, out);
}